// ChamferLoss_1614907703683
// MI455X (gfx1250) — compile-verified
//
#include <hip/hip_runtime.h>
#include <math.h>

typedef __attribute__((ext_vector_type(2))) float v2f;
typedef __attribute__((ext_vector_type(8))) float v8f;

#define B_SZ        32
#define NPTS        4096
#define WG_THREADS  256
#define ROWS_PER_WAVE 16
#define ROWS_PER_WG   128   // 8 waves * 16 rows

// One half-pass of Chamfer: for each query point, min distance to all ref
// points; block-reduced sum of sqrt(min d^2) is atomically added (scaled).
__global__ __launch_bounds__(WG_THREADS) void chamfer_halfpass(
    const float* __restrict__ query, const float* __restrict__ ref,
    float* __restrict__ out, float scale)
{
    __shared__ float bsum;
    if (threadIdx.x == 0) bsum = 0.0f;
    __syncthreads();

    const int lane = threadIdx.x & 31;
    const int wave = threadIdx.x >> 5;
    const int l    = lane & 15;     // column within half (N for B/C, M for A)
    const int half = lane >> 4;

    const int tilesPerBatch = NPTS / ROWS_PER_WG;          // 32
    const int b    = blockIdx.x / tilesPerBatch;
    const int tile = blockIdx.x % tilesPerBatch;
    const int m0   = tile * ROWS_PER_WG + wave * ROWS_PER_WAVE;

    // ---- A matrix (16x4 f32): rows m0..m0+15 of query, K = {x,y,z,0} ----
    // Layout: lanes 0-15 hold K=0(v0),K=1(v1); lanes 16-31 hold K=2(v0),K=3(v1)
    const float* qp = query + (size_t)(b * NPTS + m0 + l) * 3;
    const float qx = qp[0], qy = qp[1], qz = qp[2];
    v2f A;
    A.x = half ? qz : qx;
    A.y = half ? 0.0f : qy;
    const float q2 = qx * qx + qy * qy + qz * qz;  // |q|^2 for m = m0 + l

    // Running min of (r^2 - 2*cross) per C register (row m = i + 8*half)
    float acc[8];
#pragma unroll
    for (int i = 0; i < 8; ++i) acc[i] = 3.402823466e+38f;

    const float* rbase = ref + (size_t)b * NPTS * 3;

#pragma unroll 2
    for (int n0 = 0; n0 < NPTS; n0 += 16) {
        // ---- B matrix (4x16 f32): ref tile n0..n0+15, K = {x,y,z,0} ----
        const float* rp = rbase + (size_t)(n0 + l) * 3;
        const float rx = rp[0], ry = rp[1], rz = rp[2];
        v2f Bm;
        Bm.x = half ? rz : rx;
        Bm.y = half ? 0.0f : ry;
        const float r2 = rx * rx + ry * ry + rz * rz;  // |r|^2 for n = n0 + l

        // cross[m][n] = q_m . r_n  via matrix core (C = 0)
        v8f c = {};
        c = __builtin_amdgcn_wmma_f32_16x16x4_f32(
            /*neg_a=*/false, A, /*neg_b=*/false, Bm,
            /*c_mod=*/(short)0, c, /*reuse_a=*/false, /*reuse_b=*/false);

        // C layout: lane holds N = n0 + l; VGPR i holds M = m0 + i + 8*half
#pragma unroll
        for (int i = 0; i < 8; ++i) {
            float cand = fmaf(-2.0f, c[i], r2);   // r^2 - 2*cross
            acc[i] = fminf(acc[i], cand);
        }
    }

    // Min-reduce across the 16 lanes (N dimension) within each half.
#pragma unroll
    for (int i = 0; i < 8; ++i) {
#pragma unroll
        for (int mask = 1; mask < 16; mask <<= 1)
            acc[i] = fminf(acc[i], __shfl_xor(acc[i], mask, 32));
    }

    // Row m = m0 + i (half 0) / m0 + 8 + i (half 1). q2 for row i lives in
    // lane i (half 0) or lane 8+i (same value also in lane 24+i).
    float part = 0.0f;
#pragma unroll
    for (int i = 0; i < 8; ++i) {
        int src = (lane < 16) ? i : (8 + i);
        float q2i = __shfl(q2, src, 32);
        float d2  = fmaxf(q2i + acc[i], 1e-12f);
        part += sqrtf(d2);
    }

    // Lanes 0 and 16 carry the 8-row partial sums for their half.
    if (l == 0) atomicAdd(&bsum, part);
    __syncthreads();
    if (threadIdx.x == 0) atomicAdd(out, bsum * scale);
}

__global__ void chamfer_zero(float* out) { out[0] = 0.0f; }

extern "C" void kernel_launch(void* const* d_in, const int* in_sizes, int n_in,
                              void* d_out, int out_size, void* d_ws, size_t ws_size,
                              hipStream_t stream) {
    (void)in_sizes; (void)n_in; (void)out_size; (void)d_ws; (void)ws_size;
    const float* pred = (const float*)d_in[0];
    const float* gt   = (const float*)d_in[1];
    float* out = (float*)d_out;

    chamfer_zero<<<1, 1, 0, stream>>>(out);

    const float scale = 1.0f / (float)(B_SZ * NPTS);   // mean over B*M / B*N
    dim3 grid(B_SZ * (NPTS / ROWS_PER_WG));            // 1024 blocks
    dim3 block(WG_THREADS);

    // chamfer2: mean over pred points of min over gt  (query=pred, ref=gt)
    chamfer_halfpass<<<grid, block, 0, stream>>>(pred, gt, out, scale);
    // chamfer1: mean over gt points of min over pred  (query=gt, ref=pred)
    chamfer_halfpass<<<grid, block, 0, stream>>>(gt, pred, out, scale);
}